// RelationEncoder_6124623364293
// MI455X (gfx1250) — compile-verified
//
#include <hip/hip_runtime.h>

// ---------------------------------------------------------------------------
// GATConv (HEADS=2, HID=128, D=128) for gfx1250 (MI455X).
// fp32 WMMA (V_WMMA_F32_16X16X4_F32) for both GEMMs; atomic-based segment
// softmax + scatter for the edge phase.
// ---------------------------------------------------------------------------

typedef float v2f __attribute__((ext_vector_type(2)));
typedef float v8f __attribute__((ext_vector_type(8)));

#define HEADS 2
#define HID   128
#define DIN   128
#define HC    (HEADS * HID)   // 256

__device__ __forceinline__ float lrelu(float x, float s) {
    return x > 0.0f ? x : s * x;
}

// float atomic max via int punning (amax initialized to -inf)
__device__ __forceinline__ void atomicMaxFloat(float* addr, float v) {
    if (v >= 0.0f)
        atomicMax((int*)addr, __float_as_int(v));
    else
        atomicMin((unsigned int*)addr, __float_as_uint(v));
}

// ---------------------------------------------------------------------------
// 0. init workspace: agg = 0, amax = -inf, denom = 0, meansum = 0
// ---------------------------------------------------------------------------
__global__ void init_ws_kernel(float* __restrict__ agg, float* __restrict__ amax,
                               float* __restrict__ denom, float* __restrict__ meansum,
                               int n) {
    int total = n * HC;
    for (int i = blockIdx.x * blockDim.x + threadIdx.x; i < total;
         i += gridDim.x * blockDim.x) {
        agg[i] = 0.0f;
        if (i < n * HEADS) {
            amax[i]  = -__builtin_inff();
            denom[i] = 0.0f;
        }
        if (i == 0) meansum[0] = 0.0f;
    }
}

// ---------------------------------------------------------------------------
// 1. sum(edge_attr) -> meansum (divide by E at use site)
// ---------------------------------------------------------------------------
__global__ void edge_attr_sum_kernel(const float* __restrict__ ea,
                                     float* __restrict__ meansum, int E) {
    __shared__ float sd[256];
    float acc = 0.0f;
    for (int i = blockIdx.x * blockDim.x + threadIdx.x; i < E;
         i += gridDim.x * blockDim.x)
        acc += ea[i];
    sd[threadIdx.x] = acc;
    __syncthreads();
    for (int s = 128; s > 0; s >>= 1) {
        if ((int)threadIdx.x < s) sd[threadIdx.x] += sd[threadIdx.x + s];
        __syncthreads();
    }
    if (threadIdx.x == 0) atomicAdd(meansum, sd[0]);
}

// ---------------------------------------------------------------------------
// 2. coef[h] = dot(W_edge[0, h*HID:(h+1)*HID], att_edge[h, :])
// ---------------------------------------------------------------------------
__global__ void coef_kernel(const float* __restrict__ W_edge,
                            const float* __restrict__ att_edge,
                            float* __restrict__ coef) {
    __shared__ float sd[256];
    int t = threadIdx.x;                  // 0..255: h = t/128, c = t%128
    sd[t] = W_edge[t] * att_edge[t];
    __syncthreads();
    for (int s = 64; s > 0; s >>= 1) {
        if ((t & 127) < s) sd[t] += sd[t + s];
        __syncthreads();
    }
    if (t == 0)   coef[0] = sd[0];
    if (t == 128) coef[1] = sd[128];
}

// ---------------------------------------------------------------------------
// 3. GEMM1: H[M, 256] = X[M, 128] @ W_gat[128, 256]   (fp32 WMMA 16x16x4)
//    grid = (M/16, 256/(16*blockDim.y)), block = (32, 4). One wave = one
//    16x16 C tile. EXEC all-ones (no divergence).
// ---------------------------------------------------------------------------
__global__ void gemm1_wmma_kernel(const float* __restrict__ X,
                                  const float* __restrict__ W,
                                  float* __restrict__ H) {
    const int K = DIN, Nc = HC;
    int lane = threadIdx.x;                                  // 0..31
    int nt   = blockIdx.y * blockDim.y + threadIdx.y;        // N tile id
    int mt   = blockIdx.x;                                   // M tile id
    int m    = mt * 16 + (lane & 15);
    int ncol = nt * 16 + (lane & 15);
    int kk   = (lane >> 4) << 1;                             // 0 or 2

    v8f acc = {};
    for (int k0 = 0; k0 < K; k0 += 4) {
        v2f a, b;
        // A 16x4: VGPR0 = K {0|2}, VGPR1 = K {1|3}, lane-half selects pair
        a[0] = X[(size_t)m * K + k0 + kk];
        a[1] = X[(size_t)m * K + k0 + kk + 1];
        // B 4x16: rows striped across lane halves, N = lane&15
        b[0] = W[(size_t)(k0 + kk) * Nc + ncol];
        b[1] = W[(size_t)(k0 + kk + 1) * Nc + ncol];
        acc = __builtin_amdgcn_wmma_f32_16x16x4_f32(
            false, a, false, b, (short)0, acc, false, false);
    }
    // C/D: VGPR v -> M = v (+8 for upper lane half), N = lane&15
    int mb = mt * 16 + ((lane >> 4) << 3);
#pragma unroll
    for (int v = 0; v < 8; ++v)
        H[(size_t)(mb + v) * Nc + ncol] = acc[v];
}

// ---------------------------------------------------------------------------
// 4. a_src[n,h] = dot(h[n,h,:], att_src[h,:]); same for a_dst. Wave per node.
// ---------------------------------------------------------------------------
__global__ void attn_dots_kernel(const float* __restrict__ H,
                                 const float* __restrict__ att_src,
                                 const float* __restrict__ att_dst,
                                 float* __restrict__ ASRC,
                                 float* __restrict__ ADST, int Nn) {
    int wid  = (int)((blockIdx.x * blockDim.x + threadIdx.x) >> 5);
    int lane = threadIdx.x & 31;
    if (wid >= Nn) return;
    const float4* hn = (const float4*)(H + (size_t)wid * HC);
    const float4* as = (const float4*)att_src;   // 2*128 floats = 64 float4
    const float4* ad = (const float4*)att_dst;

    float4 h0 = hn[lane], h1 = hn[32 + lane];
    float4 s0v = as[lane], s1v = as[32 + lane];
    float4 d0v = ad[lane], d1v = ad[32 + lane];

    float s0 = h0.x * s0v.x + h0.y * s0v.y + h0.z * s0v.z + h0.w * s0v.w;
    float s1 = h1.x * s1v.x + h1.y * s1v.y + h1.z * s1v.z + h1.w * s1v.w;
    float d0 = h0.x * d0v.x + h0.y * d0v.y + h0.z * d0v.z + h0.w * d0v.w;
    float d1 = h1.x * d1v.x + h1.y * d1v.y + h1.z * d1v.z + h1.w * d1v.w;

#pragma unroll
    for (int msk = 16; msk >= 1; msk >>= 1) {
        s0 += __shfl_xor(s0, msk, 32);
        s1 += __shfl_xor(s1, msk, 32);
        d0 += __shfl_xor(d0, msk, 32);
        d1 += __shfl_xor(d1, msk, 32);
    }
    if (lane == 0) {
        ASRC[wid * 2 + 0] = s0;
        ASRC[wid * 2 + 1] = s1;
        ADST[wid * 2 + 0] = d0;
        ADST[wid * 2 + 1] = d1;
    }
}

// ---------------------------------------------------------------------------
// 5. per-edge logits + segment max (self loops appended: e >= E -> loop e-E)
// ---------------------------------------------------------------------------
__global__ void edge_logits_kernel(const int* __restrict__ ei,
                                   const float* __restrict__ ea,
                                   const float* __restrict__ ASRC,
                                   const float* __restrict__ ADST,
                                   const float* __restrict__ coef,
                                   const float* __restrict__ meansum,
                                   float* __restrict__ ALPHA,
                                   float* __restrict__ AMAX, int E, int Nn) {
    int tot = E + Nn;
    for (int e = blockIdx.x * blockDim.x + threadIdx.x; e < tot;
         e += gridDim.x * blockDim.x) {
        int s, d;
        float av;
        if (e < E) { s = ei[e]; d = ei[E + e]; av = ea[e]; }
        else       { s = d = e - E;            av = meansum[0] / (float)E; }
#pragma unroll
        for (int h = 0; h < HEADS; ++h) {
            float al = ASRC[s * 2 + h] + ADST[d * 2 + h] + coef[h] * av;
            al = lrelu(al, 0.2f);
            ALPHA[(size_t)e * 2 + h] = al;
            atomicMaxFloat(&AMAX[d * 2 + h], al);
        }
    }
}

// ---------------------------------------------------------------------------
// 6. ex = exp(alpha - amax[dst]); denom[dst] += ex (in-place into ALPHA)
// ---------------------------------------------------------------------------
__global__ void edge_exp_kernel(const int* __restrict__ ei,
                                float* __restrict__ ALPHA,
                                const float* __restrict__ AMAX,
                                float* __restrict__ DENOM, int E, int Nn) {
    int tot = E + Nn;
    for (int e = blockIdx.x * blockDim.x + threadIdx.x; e < tot;
         e += gridDim.x * blockDim.x) {
        int d = (e < E) ? ei[E + e] : (e - E);
#pragma unroll
        for (int h = 0; h < HEADS; ++h) {
            float ex = __expf(ALPHA[(size_t)e * 2 + h] - AMAX[d * 2 + h]);
            ALPHA[(size_t)e * 2 + h] = ex;
            atomicAdd(&DENOM[d * 2 + h], ex);
        }
    }
}

// ---------------------------------------------------------------------------
// 7. scatter: agg[dst] += h[src] * w. One wave per edge; float4 gathers,
//    per-element f32 global atomics (GLOBAL_ATOMIC_ADD_F32).
// ---------------------------------------------------------------------------
__global__ void edge_scatter_kernel(const int* __restrict__ ei,
                                    const float* __restrict__ H,
                                    const float* __restrict__ EX,
                                    const float* __restrict__ DENOM,
                                    float* __restrict__ AGG, int E, int Nn) {
    int tot  = E + Nn;
    int wid  = (int)(blockIdx.x * (blockDim.x >> 5) + (threadIdx.x >> 5));
    int lane = threadIdx.x & 31;
    if (wid >= tot) return;
    int s, d;
    if (wid < E) { s = ei[wid]; d = ei[E + wid]; }
    else         { s = d = wid - E; }

    float w0 = EX[(size_t)wid * 2 + 0] / DENOM[d * 2 + 0];
    float w1 = EX[(size_t)wid * 2 + 1] / DENOM[d * 2 + 1];

    const float4* hs = (const float4*)(H + (size_t)s * HC);   // 64 float4
    float4 v0 = hs[lane];          // head 0, channels lane*4..+3
    float4 v1 = hs[32 + lane];     // head 1
    float* b0 = AGG + (size_t)d * HC + lane * 4;
    float* b1 = b0 + HID;
    atomicAdd(b0 + 0, v0.x * w0);
    atomicAdd(b0 + 1, v0.y * w0);
    atomicAdd(b0 + 2, v0.z * w0);
    atomicAdd(b0 + 3, v0.w * w0);
    atomicAdd(b1 + 0, v1.x * w1);
    atomicAdd(b1 + 1, v1.y * w1);
    atomicAdd(b1 + 2, v1.z * w1);
    atomicAdd(b1 + 3, v1.w * w1);
}

// ---------------------------------------------------------------------------
// 8. GEMM2: OUT[M,128] = lrelu(AGG + b_gat, 0.01) @ W_lin[256,128] + b_lin
//    Fused epilogue on the A operand; fp32 WMMA 16x16x4.
// ---------------------------------------------------------------------------
__global__ void gemm2_wmma_kernel(const float* __restrict__ AGG,
                                  const float* __restrict__ bgat,
                                  const float* __restrict__ Wl,
                                  const float* __restrict__ bl,
                                  float* __restrict__ OUT) {
    const int K = HC, Nc = HID;
    int lane = threadIdx.x;
    int nt   = blockIdx.y * blockDim.y + threadIdx.y;
    int mt   = blockIdx.x;
    int m    = mt * 16 + (lane & 15);
    int ncol = nt * 16 + (lane & 15);
    int kk   = (lane >> 4) << 1;

    v8f acc = {};
    for (int k0 = 0; k0 < K; k0 += 4) {
        v2f a, b;
        int ka = k0 + kk;
        a[0] = lrelu(AGG[(size_t)m * K + ka]     + bgat[ka],     0.01f);
        a[1] = lrelu(AGG[(size_t)m * K + ka + 1] + bgat[ka + 1], 0.01f);
        b[0] = Wl[(size_t)ka * Nc + ncol];
        b[1] = Wl[(size_t)(ka + 1) * Nc + ncol];
        acc = __builtin_amdgcn_wmma_f32_16x16x4_f32(
            false, a, false, b, (short)0, acc, false, false);
    }
    int mb = mt * 16 + ((lane >> 4) << 3);
    float bias = bl[ncol];
#pragma unroll
    for (int v = 0; v < 8; ++v)
        OUT[(size_t)(mb + v) * Nc + ncol] = acc[v] + bias;
}

// ---------------------------------------------------------------------------
// launcher
// ---------------------------------------------------------------------------
extern "C" void kernel_launch(void* const* d_in, const int* in_sizes, int n_in,
                              void* d_out, int out_size, void* d_ws, size_t ws_size,
                              hipStream_t stream) {
    const float* x        = (const float*)d_in[0];   // [N,128]
    const int*   eidx     = (const int*)  d_in[1];   // [2,E]
    const float* eattr    = (const float*)d_in[2];   // [E,1]
    const float* W_gat    = (const float*)d_in[3];   // [128,256]
    const float* b_gat    = (const float*)d_in[4];   // [256]
    const float* att_src  = (const float*)d_in[5];   // [2,128]
    const float* att_dst  = (const float*)d_in[6];   // [2,128]
    const float* W_edge   = (const float*)d_in[7];   // [1,256]
    const float* att_edge = (const float*)d_in[8];   // [2,128]
    const float* W_lin    = (const float*)d_in[9];   // [256,128]
    const float* b_lin    = (const float*)d_in[10];  // [128]
    float*       out      = (float*)d_out;

    const int N = in_sizes[0] / DIN;   // 32768
    const int E = in_sizes[2];         // 524288
    const int TOT = E + N;

    float* ws = (float*)d_ws;
    size_t off = 0;
    float* H     = ws + off; off += (size_t)N * HC;
    float* AGG   = ws + off; off += (size_t)N * HC;
    float* ASRC  = ws + off; off += (size_t)N * HEADS;
    float* ADST  = ws + off; off += (size_t)N * HEADS;
    float* AMAX  = ws + off; off += (size_t)N * HEADS;
    float* DENOM = ws + off; off += (size_t)N * HEADS;
    float* ALPHA = ws + off; off += (size_t)TOT * HEADS;
    float* COEF  = ws + off; off += 2;
    float* MEAN  = ws + off; off += 2;

    // 0. init
    init_ws_kernel<<<2048, 256, 0, stream>>>(AGG, AMAX, DENOM, MEAN, N);
    // 1. edge_attr mean (sum)
    edge_attr_sum_kernel<<<512, 256, 0, stream>>>(eattr, MEAN, E);
    // 2. per-head edge coefficient
    coef_kernel<<<1, 256, 0, stream>>>(W_edge, att_edge, COEF);
    // 3. H = X @ W_gat   (WMMA fp32)
    gemm1_wmma_kernel<<<dim3(N / 16, HC / 64), dim3(32, 4), 0, stream>>>(x, W_gat, H);
    // 4. attention dots
    attn_dots_kernel<<<(N * 32 + 255) / 256, 256, 0, stream>>>(
        H, att_src, att_dst, ASRC, ADST, N);
    // 5. logits + segment max
    edge_logits_kernel<<<(TOT + 255) / 256, 256, 0, stream>>>(
        eidx, eattr, ASRC, ADST, COEF, MEAN, ALPHA, AMAX, E, N);
    // 6. exp + segment sum
    edge_exp_kernel<<<(TOT + 255) / 256, 256, 0, stream>>>(
        eidx, ALPHA, AMAX, DENOM, E, N);
    // 7. weighted scatter
    edge_scatter_kernel<<<(TOT + 7) / 8, 256, 0, stream>>>(
        eidx, H, ALPHA, DENOM, AGG, E, N);
    // 8. OUT = lrelu(AGG + b_gat) @ W_lin + b_lin   (WMMA fp32)
    gemm2_wmma_kernel<<<dim3(N / 16, HID / 64), dim3(32, 4), 0, stream>>>(
        AGG, b_gat, W_lin, b_lin, out);
}